// MessagePassingConvolution_58926951301691
// MI455X (gfx1250) — compile-verified
//
#include <hip/hip_runtime.h>
#include <hip/hip_bf16.h>
#include <math.h>

typedef __attribute__((ext_vector_type(2))) float v2f;
typedef __attribute__((ext_vector_type(8))) float v8f;

#define INV_SQRT3 0.57735026919f

static __device__ __forceinline__ v8f wmma_f32_k4(v2f a, v2f b, v8f c) {
    // D = A(16x4 f32) * B(4x16 f32) + C(16x16 f32)
    return __builtin_amdgcn_wmma_f32_16x16x4_f32(
        /*neg_a=*/false, a, /*neg_b=*/false, b,
        /*c_mod=*/(short)0, c, /*reuse_a=*/false, /*reuse_b=*/false);
}

static __device__ __forceinline__ float silu(float x) {
    return x / (1.0f + __expf(-x));
}

__global__ void zero_kernel(float* __restrict__ out, int n) {
    int i = blockIdx.x * blockDim.x + threadIdx.x;
    if (i < n) out[i] = 0.0f;
}

// One workgroup (4 waves / 128 threads) handles a tile of 16 edges.
__global__ __launch_bounds__(128) void mpc_kernel(
    const float* __restrict__ edge_s,   // (E,64)
    const float* __restrict__ edge_v,   // (E,64,3)
    const float* __restrict__ attr_s,   // (E,1)
    const float* __restrict__ attr_v,   // (E,3)
    const float* __restrict__ W1,       // (64,128)
    const float* __restrict__ b1,       // (128)
    const float* __restrict__ W2,       // (128,128)
    const float* __restrict__ b2,       // (128)
    const float* __restrict__ Wg,       // (128,256)
    const float* __restrict__ bg,       // (256)
    const int*   __restrict__ receivers,// (E)
    float*       __restrict__ out)      // (N_NODES,512)
{
    // LDS staging; row stride padded (+4 words) so the A-fragment loads
    // (16 lanes reading one K-column pair across rows) hit distinct banks.
    __shared__ float h1s[16 * 132];     // silu(edge_s@W1+b1)
    __shared__ float h2s[16 * 132];     // silu(h1@W2+b2)
    __shared__ float gts[16 * 260];     // h2@Wg+bg

    const int tile  = blockIdx.x;
    const int e0    = tile * 16;
    const int lane  = threadIdx.x & 31;
    const int wave  = threadIdx.x >> 5;
    const int row   = lane & 15;   // matrix row (A) / column (B,C) owned by lane
    const int khalf = lane >> 4;   // 0 -> K=k0,k0+1 ; 1 -> K=k0+2,k0+3

    // Warm edge_v for the epilogue while the MLP runs (global_prefetch_b8).
    {
        int e = threadIdx.x >> 3;
        int q = threadIdx.x & 7;
        __builtin_prefetch(edge_v + (size_t)(e0 + e) * 192 + q * 24, 0, 0);
    }

    // ---------------- GEMM1: h1 = silu(edge_s @ W1 + b1), 16x128 ----------
    {
        const int n0 = wave * 32;                       // two N-tiles per wave
        v8f acc0 = {}, acc1 = {};
        const float* As = edge_s + (size_t)(e0 + row) * 64 + 2 * khalf;
#pragma unroll 4
        for (int k0 = 0; k0 < 64; k0 += 4) {
            v2f a = *(const v2f*)(As + k0);             // K=k0+2*khalf, +1
            const float* B0 = W1 + (size_t)(k0 + 2 * khalf) * 128 + n0 + row;
            v2f bb0; bb0.x = B0[0];  bb0.y = B0[128];
            v2f bb1; bb1.x = B0[16]; bb1.y = B0[128 + 16];
            acc0 = wmma_f32_k4(a, bb0, acc0);
            acc1 = wmma_f32_k4(a, bb1, acc1);
        }
        float bias0 = b1[n0 + row];
        float bias1 = b1[n0 + 16 + row];
#pragma unroll
        for (int j = 0; j < 8; ++j) {
            int m = j + 8 * khalf;                      // edge row in tile
            h1s[m * 132 + n0 + row]      = silu(acc0[j] + bias0);
            h1s[m * 132 + n0 + 16 + row] = silu(acc1[j] + bias1);
        }
    }
    __syncthreads();

    // ---------------- GEMM2: h2 = silu(h1 @ W2 + b2), 16x128 --------------
    {
        const int n0 = wave * 32;
        v8f acc0 = {}, acc1 = {};
        const float* As = h1s + row * 132 + 2 * khalf;
#pragma unroll 4
        for (int k0 = 0; k0 < 128; k0 += 4) {
            v2f a = *(const v2f*)(As + k0);
            const float* B0 = W2 + (size_t)(k0 + 2 * khalf) * 128 + n0 + row;
            v2f bb0; bb0.x = B0[0];  bb0.y = B0[128];
            v2f bb1; bb1.x = B0[16]; bb1.y = B0[128 + 16];
            acc0 = wmma_f32_k4(a, bb0, acc0);
            acc1 = wmma_f32_k4(a, bb1, acc1);
        }
        float bias0 = b2[n0 + row];
        float bias1 = b2[n0 + 16 + row];
#pragma unroll
        for (int j = 0; j < 8; ++j) {
            int m = j + 8 * khalf;
            h2s[m * 132 + n0 + row]      = silu(acc0[j] + bias0);
            h2s[m * 132 + n0 + 16 + row] = silu(acc1[j] + bias1);
        }
    }
    __syncthreads();

    // ---------------- GEMM3: gate = h2 @ Wg + bg, 16x256 ------------------
    {
        const int n0 = wave * 64;                       // four N-tiles per wave
        v8f acc[4] = {{}, {}, {}, {}};
        const float* As = h2s + row * 132 + 2 * khalf;
#pragma unroll 2
        for (int k0 = 0; k0 < 128; k0 += 4) {
            v2f a = *(const v2f*)(As + k0);
            const float* B0 = Wg + (size_t)(k0 + 2 * khalf) * 256 + n0 + row;
#pragma unroll
            for (int t = 0; t < 4; ++t) {
                v2f bb; bb.x = B0[16 * t]; bb.y = B0[256 + 16 * t];
                acc[t] = wmma_f32_k4(a, bb, acc[t]);
            }
        }
#pragma unroll
        for (int t = 0; t < 4; ++t) {
            int col = n0 + 16 * t + row;
            float bias = bg[col];
#pragma unroll
            for (int j = 0; j < 8; ++j) {
                int m = j + 8 * khalf;
                gts[m * 260 + col] = acc[t][j] + bias;
            }
        }
    }
    __syncthreads();

    // ---------------- Epilogue: gate messages + scatter-add ---------------
    // 8 threads per edge; thread handles 64 contiguous output columns of 512.
    {
        const int   e    = threadIdx.x >> 3;
        const int   c0   = (threadIdx.x & 7) * 64;
        const int   ge   = e0 + e;
        const float* es  = edge_s + (size_t)ge * 64;
        const float* ev  = edge_v + (size_t)ge * 192;
        const float  as  = attr_s[ge];
        const float  av0 = attr_v[ge * 3 + 0];
        const float  av1 = attr_v[ge * 3 + 1];
        const float  av2 = attr_v[ge * 3 + 2];
        const float* g   = gts + e * 260;               // 256 gate values
        float* dst = out + (size_t)receivers[ge] * 512;
        const float SC = 0.25f;                         // 1/sqrt(16)

        if (c0 == 0) {
            // m0 first half: edge_s * attr_s * g0
#pragma unroll 4
            for (int f = 0; f < 64; ++f) {
                atomicAdd(dst + f, es[f] * as * g[f] * SC);
            }
        } else if (c0 == 64) {
            // m0 second half: dot(edge_v[f], attr_v) * INV_SQRT3 * g0
#pragma unroll 4
            for (int f = 0; f < 64; ++f) {
                float d = ev[f * 3] * av0 + ev[f * 3 + 1] * av1 + ev[f * 3 + 2] * av2;
                atomicAdd(dst + 64 + f, d * INV_SQRT3 * g[64 + f] * SC);
            }
        } else {
            // m1 flattened (128 features x 3 comps), gated by g1 = g[128+f]
            const int j0 = c0 - 128;                    // 0,64,...,320
#pragma unroll 4
            for (int t = 0; t < 64; ++t) {
                int j    = j0 + t;
                int f    = j / 3;
                int comp = j - 3 * f;
                float a  = (comp == 0) ? av0 : ((comp == 1) ? av1 : av2);
                float m1 = (f < 64) ? es[f] * a : ev[(f - 64) * 3 + comp] * as;
                atomicAdd(dst + 128 + j, m1 * g[128 + f] * SC);
            }
        }
    }
}

extern "C" void kernel_launch(void* const* d_in, const int* in_sizes, int n_in,
                              void* d_out, int out_size, void* d_ws, size_t ws_size,
                              hipStream_t stream) {
    const float* edge_s    = (const float*)d_in[0];
    const float* edge_v    = (const float*)d_in[1];
    const float* attr_s    = (const float*)d_in[2];
    const float* attr_v    = (const float*)d_in[3];
    const float* W1        = (const float*)d_in[4];
    const float* b1        = (const float*)d_in[5];
    const float* W2        = (const float*)d_in[6];
    const float* b2        = (const float*)d_in[7];
    const float* Wg        = (const float*)d_in[8];
    const float* bg        = (const float*)d_in[9];
    const int*   receivers = (const int*)d_in[10];
    float* out = (float*)d_out;

    const int E = in_sizes[10];          // 160000 edges
    const int tiles = E / 16;            // 10000 workgroups

    zero_kernel<<<(out_size + 255) / 256, 256, 0, stream>>>(out, out_size);
    mpc_kernel<<<tiles, 128, 0, stream>>>(edge_s, edge_v, attr_s, attr_v,
                                          W1, b1, W2, b2, Wg, bg,
                                          receivers, out);
}